// NetTGCNBasic_67070209295120
// MI455X (gfx1250) — compile-verified
//
#include <hip/hip_runtime.h>
#include <hip/hip_bf16.h>

#define N_NODES 10000
#define N_EDGES 160000
#define KCHEB 25
#define HLEN 12
#define G1 32
#define NUM_CLASSES 10
#define BATCH 32
#define ROWF (BATCH * HLEN)          // 384 floats per node row in tx layout [N, B*H]

typedef __attribute__((ext_vector_type(2))) float v2f;
typedef __attribute__((ext_vector_type(8))) float v8f;

// ---------------- utility kernels ----------------

__global__ void zero_f32(float* p, long n) {
    long i = (long)blockIdx.x * blockDim.x + threadIdx.x;
    if (i < n) p[i] = 0.0f;
}

__global__ void count_deg(const int* __restrict__ row, float* deg) {
    int e = blockIdx.x * blockDim.x + threadIdx.x;
    if (e < N_EDGES) atomicAdd(&deg[row[e]], 1.0f);
}

__global__ void make_dinv(const float* __restrict__ deg, float* __restrict__ dinv) {
    int n = blockIdx.x * blockDim.x + threadIdx.x;
    if (n < N_NODES) {
        float d = deg[n];
        dinv[n] = (d > 0.0f) ? rsqrtf(d) : 0.0f;
    }
}

__global__ void make_norm(const int* __restrict__ row, const int* __restrict__ col,
                          const float* __restrict__ dinv, float* __restrict__ norm) {
    int e = blockIdx.x * blockDim.x + threadIdx.x;
    if (e < N_EDGES) norm[e] = -dinv[row[e]] * dinv[col[e]];
}

// xf[(n*B + b)*H + h'] = sum_t x[(b*N + n)*H + t] * cos(2*pi*h'*t/12)
// tx state layout: [N, B, H] so each node's (B*H)=384 floats are contiguous.
__global__ void fft_real(const float* __restrict__ x, float* __restrict__ xf) {
    int idx = blockIdx.x * blockDim.x + threadIdx.x;   // = b*N + n (coalesced reads)
    if (idx >= BATCH * N_NODES) return;
    const float ctab[12] = {
        1.0f, 0.8660254037844387f, 0.5f, 0.0f, -0.5f, -0.8660254037844387f,
        -1.0f, -0.8660254037844387f, -0.5f, 0.0f, 0.5f, 0.8660254037844387f };
    int b = idx / N_NODES;
    int n = idx - b * N_NODES;
    const float* xr = x + (size_t)idx * HLEN;
    float v[HLEN];
#pragma unroll
    for (int t = 0; t < HLEN; ++t) v[t] = xr[t];
    float* o = xf + ((size_t)n * BATCH + b) * HLEN;
#pragma unroll
    for (int hh = 0; hh < HLEN; ++hh) {
        float s = 0.0f;
#pragma unroll
        for (int t = 0; t < HLEN; ++t) s += v[t] * ctab[(hh * t) % 12];
        o[hh] = s;
    }
}

// agg[row[e]][*] += norm[e] * z[col[e]][*]   (one WAVE per edge; 384 floats/node)
// Gather: 3x float4 per lane (coalesced 512B per instruction).
// Scatter: 12 scalar float atomics per lane into one contiguous 1536B span.
__global__ void prop_edges(const float* __restrict__ z, const int* __restrict__ row,
                           const int* __restrict__ col, const float* __restrict__ norm,
                           float* agg) {
    int wid = (blockIdx.x * blockDim.x + threadIdx.x) >> 5;
    int lane = threadIdx.x & 31;
    if (wid >= N_EDGES) return;
    int r = row[wid], c = col[wid];
    float w = norm[wid];
    const float4* src = (const float4*)(z + (size_t)c * ROWF);
    float* dst = agg + (size_t)r * ROWF;
#pragma unroll
    for (int q = 0; q < 3; ++q) {
        float4 v = src[q * 32 + lane];
        int base = (q * 32 + lane) * 4;
        atomicAdd(&dst[base + 0], w * v.x);
        atomicAdd(&dst[base + 1], w * v.y);
        atomicAdd(&dst[base + 2], w * v.z);
        atomicAdd(&dst[base + 3], w * v.w);
    }
}

// tout = alpha*agg + beta*tin   (tin/tout may alias)
__global__ void combine(const float* __restrict__ agg, const float* tin,
                        float* tout, float alpha, float beta, long n) {
    long i = (long)blockIdx.x * blockDim.x + threadIdx.x;
    if (i < n) tout[i] = alpha * agg[i] + beta * tin[i];
}

// ---------------- WMMA einsum: out[M,32] += tx[M,12] @ W[12,32] ----------------
// M = N*B = 320000 rows (row r = n*32 + b), 20000 full 16-row tiles -> EXEC all-1.
// Per wave: two 16-col tiles, K=12 as 3 chained V_WMMA_F32_16X16X4_F32.
__global__ void cheb_einsum_wmma(const float* __restrict__ tx,
                                 const float* __restrict__ W,
                                 float* __restrict__ out) {
    const int M = BATCH * N_NODES;
    int wave = (blockIdx.x * blockDim.x + threadIdx.x) >> 5;
    int lane = threadIdx.x & 31;
    int r0 = wave * 16;
    if (r0 >= M) return;
    int lhalf = lane >> 4;      // 0: K={0,1}, 1: K={2,3} halves
    int l16   = lane & 15;

    // A fragment: lanes 0-15 hold (K=ks+0, ks+1), lanes 16-31 hold (ks+2, ks+3)
    const float* arow = tx + ((size_t)(r0 + l16)) * HLEN + lhalf * 2;
    v2f a0 = { arow[0], arow[1] };
    v2f a1 = { arow[4], arow[5] };
    v2f a2 = { arow[8], arow[9] };

#pragma unroll
    for (int t = 0; t < 2; ++t) {           // two 16-wide column tiles of G1=32
        int f = t * 16 + l16;
        int kb = lhalf * 2;
        v2f b0 = { W[(kb + 0) * G1 + f], W[(kb + 1) * G1 + f] };
        v2f b1 = { W[(kb + 4) * G1 + f], W[(kb + 5) * G1 + f] };
        v2f b2 = { W[(kb + 8) * G1 + f], W[(kb + 9) * G1 + f] };

        // C/D layout: VGPR g -> row r0 + g + 8*lhalf, col f
        float* op = out + ((size_t)(r0 + lhalf * 8)) * G1 + f;
        v8f c;
#pragma unroll
        for (int g = 0; g < 8; ++g) c[g] = op[(size_t)g * G1];

        c = __builtin_amdgcn_wmma_f32_16x16x4_f32(false, a0, false, b0, (short)0, c, false, false);
        c = __builtin_amdgcn_wmma_f32_16x16x4_f32(false, a1, false, b1, (short)0, c, false, false);
        c = __builtin_amdgcn_wmma_f32_16x16x4_f32(false, a2, false, b2, (short)0, c, false, false);

#pragma unroll
        for (int g = 0; g < 8; ++g) op[(size_t)g * G1] = c[g];
    }
}

// ---------------- FC + bias + relu: logits[b,c] ----------------
// out layout: out[(n*32 + b)*32 + f]; reference flatten index j = n*32 + f.
__global__ void fc_reduce(const float* __restrict__ out, const float* __restrict__ conv_b,
                          const float* __restrict__ fc_w, const float* __restrict__ fc_b,
                          float* __restrict__ logits) {
    const int J = N_NODES * G1;          // 320000
    int b = blockIdx.x;
    int tid = threadIdx.x;
    float acc[NUM_CLASSES];
#pragma unroll
    for (int c = 0; c < NUM_CLASSES; ++c) acc[c] = 0.0f;
    for (int j = tid; j < J; j += blockDim.x) {
        int f = j & (G1 - 1);
        // row r = (j>>5)*32 + b  ->  element r*32 + f
        float h = out[((size_t)(j & ~(G1 - 1)) + b) * G1 + f] + conv_b[f];
        h = h > 0.0f ? h : 0.0f;
#pragma unroll
        for (int c = 0; c < NUM_CLASSES; ++c) acc[c] += h * fc_w[(size_t)c * J + j];
    }
    __shared__ float red[NUM_CLASSES][256];
#pragma unroll
    for (int c = 0; c < NUM_CLASSES; ++c) red[c][tid] = acc[c];
    __syncthreads();
    for (int s = 128; s > 0; s >>= 1) {
        if (tid < s)
#pragma unroll
            for (int c = 0; c < NUM_CLASSES; ++c) red[c][tid] += red[c][tid + s];
        __syncthreads();
    }
    if (tid < NUM_CLASSES) logits[b * NUM_CLASSES + tid] = red[tid][0] + fc_b[tid];
}

__global__ void log_softmax(const float* __restrict__ logits, float* __restrict__ outp) {
    int b = threadIdx.x;
    if (b >= BATCH) return;
    const float* l = logits + b * NUM_CLASSES;
    float m = l[0];
#pragma unroll
    for (int c = 1; c < NUM_CLASSES; ++c) m = fmaxf(m, l[c]);
    float s = 0.0f;
#pragma unroll
    for (int c = 0; c < NUM_CLASSES; ++c) s += __expf(l[c] - m);
    float lse = m + __logf(s);
#pragma unroll
    for (int c = 0; c < NUM_CLASSES; ++c) outp[b * NUM_CLASSES + c] = l[c] - lse;
}

// ---------------- host orchestration ----------------

static inline long aline(long x) { return (x + 63) & ~63L; }   // 64-elem (256B) align

extern "C" void kernel_launch(void* const* d_in, const int* in_sizes, int n_in,
                              void* d_out, int out_size, void* d_ws, size_t ws_size,
                              hipStream_t stream) {
    const float* x      = (const float*)d_in[0];
    const int*   ei     = (const int*)d_in[1];
    const float* conv_w = (const float*)d_in[2];
    const float* conv_b = (const float*)d_in[3];
    const float* fc_w   = (const float*)d_in[4];
    const float* fc_b   = (const float*)d_in[5];
    float* out = (float*)d_out;

    const int* row = ei;
    const int* col = ei + N_EDGES;

    // workspace carve (floats)
    float* w = (float*)d_ws;
    long off = 0;
    float* deg    = w + off; off = aline(off + N_NODES);
    float* dinv   = w + off; off = aline(off + N_NODES);
    float* norm   = w + off; off = aline(off + N_EDGES);
    const long TXN = (long)N_NODES * ROWF;                  // 3,840,000
    float* tx0 = w + off; off = aline(off + TXN);
    float* tx1 = w + off; off = aline(off + TXN);
    float* agg = w + off; off = aline(off + TXN);
    const long OUTN = (long)BATCH * N_NODES * G1;           // 10,240,000
    float* outbuf = w + off; off = aline(off + OUTN);
    float* logits = w + off; off = aline(off + BATCH * NUM_CLASSES);

    const int T = 256;
    auto blk = [](long n, int t) { return (int)((n + t - 1) / t); };

    // norm coefficients
    zero_f32<<<blk(N_NODES, T), T, 0, stream>>>(deg, N_NODES);
    count_deg<<<blk(N_EDGES, T), T, 0, stream>>>(row, deg);
    make_dinv<<<blk(N_NODES, T), T, 0, stream>>>(deg, dinv);
    make_norm<<<blk(N_EDGES, T), T, 0, stream>>>(row, col, dinv, norm);

    // xf = Re(FFT(x)) -> tx0 (this is T_0), layout [N, B, H]
    fft_real<<<blk((long)BATCH * N_NODES, T), T, 0, stream>>>(x, tx0);

    // out accumulator
    zero_f32<<<blk(OUTN, T), T, 0, stream>>>(outbuf, OUTN);

    // WMMA einsum: 20000 waves, 4 waves/block
    const int EIN_BLOCKS = (BATCH * N_NODES / 16) / 4;      // 5000
    const int EIN_T = 128;
    // prop: one wave per edge, 8 waves/block
    const int PROP_BLOCKS = N_EDGES / 8;                    // 20000
    const int PROP_T = 256;

    // k = 0
    cheb_einsum_wmma<<<EIN_BLOCKS, EIN_T, 0, stream>>>(tx0, conv_w + 0 * HLEN * G1, outbuf);

    // k = 1 : T_1 = L~ T_0
    zero_f32<<<blk(TXN, T), T, 0, stream>>>(agg, TXN);
    prop_edges<<<PROP_BLOCKS, PROP_T, 0, stream>>>(tx0, row, col, norm, agg);
    combine<<<blk(TXN, T), T, 0, stream>>>(agg, agg, tx1, 1.0f, 0.0f, TXN);
    cheb_einsum_wmma<<<EIN_BLOCKS, EIN_T, 0, stream>>>(tx1, conv_w + 1 * HLEN * G1, outbuf);

    // k = 2..24 : T_k = 2 L~ T_{k-1} - T_{k-2}
    float* tprev = tx0;   // T_{k-2}
    float* tcur  = tx1;   // T_{k-1}
    for (int k = 2; k < KCHEB; ++k) {
        zero_f32<<<blk(TXN, T), T, 0, stream>>>(agg, TXN);
        prop_edges<<<PROP_BLOCKS, PROP_T, 0, stream>>>(tcur, row, col, norm, agg);
        // write T_k into the tprev buffer (elementwise in-place safe)
        combine<<<blk(TXN, T), T, 0, stream>>>(agg, tprev, tprev, 2.0f, -1.0f, TXN);
        cheb_einsum_wmma<<<EIN_BLOCKS, EIN_T, 0, stream>>>(tprev, conv_w + (size_t)k * HLEN * G1, outbuf);
        float* t = tprev; tprev = tcur; tcur = t;   // rotate
    }

    // FC + log_softmax
    fc_reduce<<<BATCH, 256, 0, stream>>>(outbuf, conv_b, fc_w, fc_b, logits);
    log_softmax<<<1, 32, 0, stream>>>(logits, out);
    (void)in_sizes; (void)n_in; (void)out_size; (void)ws_size;
}